// RubiksShiftBase_91010357002302
// MI455X (gfx1250) — compile-verified
//
#include <hip/hip_runtime.h>

// Problem constants (fixed by the reference harness)
#define NB 16
#define CB 256
#define HB 112
#define WB 112
#define HWB (HB * WB)        // 12544 elements per plane (50,176 bytes)
#define CHUNKS (HWB / 4)     // 3136 float4 chunks per plane
#define THREADS 256

typedef float v4f __attribute__((ext_vector_type(4)));

__global__ __launch_bounds__(THREADS)
void rubiks_shift_kernel(const float* __restrict__ x,
                         const float* __restrict__ shift,
                         float* __restrict__ out)
{
    __shared__ float tile[HWB];                 // 50,176 B of LDS (320 KB/WGP available)

    const int plane = blockIdx.x;               // n*C + c
    const int c     = plane & (CB - 1);
    const int tid   = threadIdx.x;

    // ---- per-channel shift decomposition (uniform within the block) ----
    const float sh   = shift[c];                // height shift
    const float sw   = shift[CB + c];           // width shift
    const float flhf = floorf(sh);
    const float flwf = floorf(sw);
    const int   flh  = (int)flhf;
    const int   flw  = (int)flwf;
    const float frh  = sh - flhf;
    const float frw  = sw - flwf;
    const float wh0  = 1.0f - frh, wh1 = frh;
    const float ww0  = 1.0f - frw, ww1 = frw;

    // ---- async stage: whole plane HBM -> LDS, every byte read exactly once ----
    const float* gbase = x + (size_t)plane * HWB;      // uniform -> lives in SGPR pair

    // 32-bit LDS byte offset of the tile (addrspace(3) pointer is 32-bit)
    __attribute__((address_space(3))) float* tile_lds =
        ( __attribute__((address_space(3))) float*)tile;
    const unsigned lds0 = (unsigned)(uintptr_t)tile_lds;

    for (int q = tid; q < CHUNKS; q += THREADS) {
        unsigned ldsoff = lds0 + (unsigned)q * 16u;    // LDS dest byte address
        int      goff   = q * 16;                      // byte offset from plane base
        // CDNA5 async DMA: copies 16 bytes from global address gbase + goff to the LDS address in ldsoff, ASYNCcnt-tracked
        asm volatile("global_load_async_to_lds_b128 %0, %1, %2"
                     :: "v"(ldsoff), "v"(goff), "s"(gbase)
                     : "memory");
    }
    // wait for this wave's async transfers, then sync the workgroup
    asm volatile("s_wait_asynccnt 0" ::: "memory");
    __syncthreads();

    // ---- compute: 4 consecutive outputs per thread-iteration ----
    float* obase = out + (size_t)plane * HWB;

    for (int q = tid; q < CHUNKS; q += THREADS) {
        const int p = q * 4;                 // flat output index of first of 4
        const int i = p / WB;                // output row (112 % 4 == 0 -> same row for all 4)
        const int j = p - i * WB;            // output col of first element

        const int r0 = i + flh;
        const int r1 = r0 + 1;
        const bool ok0 = (unsigned)r0 < (unsigned)HB;
        const bool ok1 = (unsigned)r1 < (unsigned)HB;
        const float a0 = ok0 ? wh0 : 0.0f;   // fold row-validity into row weights
        const float a1 = ok1 ? wh1 : 0.0f;
        const float* row0 = &tile[(ok0 ? r0 : 0) * WB];
        const float* row1 = &tile[(ok1 ? r1 : 0) * WB];

        const int cbase = j + flw;
        float t[5];
        #pragma unroll
        for (int d = 0; d < 5; ++d) {
            const int  cc  = cbase + d;
            const bool okc = (unsigned)cc < (unsigned)WB;
            const int  ccl = okc ? cc : 0;
            const float v  = a0 * row0[ccl] + a1 * row1[ccl];   // ds_load_b32 x2 + FMA
            t[d] = okc ? v : 0.0f;                              // fold col-validity
        }

        v4f o;
        o.x = ww0 * t[0] + ww1 * t[1];
        o.y = ww0 * t[1] + ww1 * t[2];
        o.z = ww0 * t[2] + ww1 * t[3];
        o.w = ww0 * t[3] + ww1 * t[4];

        // write-once output: nontemporal 16B coalesced store
        __builtin_nontemporal_store(o, (v4f*)(obase + p));
    }
}

extern "C" void kernel_launch(void* const* d_in, const int* in_sizes, int n_in,
                              void* d_out, int out_size, void* d_ws, size_t ws_size,
                              hipStream_t stream) {
    const float* x     = (const float*)d_in[0];   // [16, 256, 112, 112] fp32
    const float* shift = (const float*)d_in[1];   // [2, 256] fp32
    float*       out   = (float*)d_out;           // [16, 256, 112, 112] fp32

    dim3 grid(NB * CB);                            // one workgroup per (n, c) plane
    dim3 block(THREADS);
    rubiks_shift_kernel<<<grid, block, 0, stream>>>(x, shift, out);
}